// STAT_Net_V2_64922725647061
// MI455X (gfx1250) — compile-verified
//
#include <hip/hip_runtime.h>
#include <hip/hip_bf16.h>

// ---------------------------------------------------------------------------
// Shapes from the reference:
//   features: (B=64, C=16, F=2048, J=25) f32
//   W2: (E=64, C=16, NJ=12), b2: (64), Wl: (5, 1024), bl: (5)
//   output: logits (64, 5) f32
// stat1/W1/b1 are dead code in the reference (result discarded) -> skipped.
//
// MI455X plan:
//  - 210 MB feature stream ~9us at 23.3 TB/s; per-column strided gather is
//    done by the Tensor Data Mover (tensor_load_to_lds, TENSORcnt) instead of
//    per-lane VMEM: D# = 1x2048 tile, tensor_dim0_stride = 25 elements.
//  - order stats via in-LDS bitonic sort (exact ranks 0/511/512/1023/1535/
//    1536/2047), moments via tree reduction.
//  - fuse GEMM (16 slices of 64x192 . 192x64 = 25 MFLOP) in fp32 WMMA
//    v_wmma_f32_16x16x4_f32 -- precision-preserving, one wave per 16x16 tile.
// ---------------------------------------------------------------------------

#define NB    64
#define NC    16
#define NF    2048
#define NJALL 25
#define NJ    12
#define NE    64
#define NCLS  5
#define KDIM  (NC * NJ)          // 192
#define NS    16                 // 8 stats x 2 signals
#define TPB   256

typedef __attribute__((ext_vector_type(2))) float v2f;
typedef __attribute__((ext_vector_type(8))) float v8f;
typedef __attribute__((ext_vector_type(4))) unsigned int v4u;
typedef __attribute__((ext_vector_type(4))) int v4i;
typedef __attribute__((ext_vector_type(8))) int v8i;

// ---------------------------------------------------------------------------
// Kernel 1: per-column order statistics + central moments.
// One block = one column (sig, b, c, j). The 2048-element column (stride
// 100 B in global) is DMA'd into LDS by the TDM, then: moments via tree
// reduction, exact order stats via in-LDS bitonic sort.
// Output layout statA[S][b][c*12+j], S = sig*8 + statIdx  (GEMM A matrix).
// ---------------------------------------------------------------------------
__global__ void __launch_bounds__(TPB)
statnet_stats_kernel(const float* __restrict__ feat, float* __restrict__ statA) {
    __shared__ float xs[NF];
    __shared__ float red[TPB];
    __shared__ float sBcast[3];          // mean, sum2, sum4

    const int tid = threadIdx.x;
    const int col = blockIdx.x;          // 0 .. 24575
    const int sig = col / (NB * NC * NJ);
    const int rem = col - sig * (NB * NC * NJ);
    const int b   = rem / (NC * NJ);
    const int cj  = rem - b * (NC * NJ); // c*12 + j
    const int c   = cj / NJ;
    const int j   = cj - c * NJ;

    const float* p = feat + ((size_t)(b * NC + c) * NF) * NJALL + (j + NJ * sig);

    // ---- TDM: strided column gather -> contiguous LDS (one DMA per block) --
    if (tid == 0) {
        const unsigned long long ga = (unsigned long long)(uintptr_t)p;
        const unsigned lds_base = (unsigned)(uintptr_t)&xs[0];

        v4u g0;
        g0.x = 1u;                                    // count=1, user descriptor
        g0.y = lds_base;                              // lds_addr (bytes)
        g0.z = (unsigned)(ga & 0xFFFFFFFFu);          // global_addr[31:0]
        g0.w = (unsigned)((ga >> 32) & 0x01FFFFFFu)   // global_addr[56:32]
             | (2u << 30);                            // type = 2 ("image")

        v8i g1;
        g1[0] = 0x00020000;                           // data_size=2 (4B), wg_mask=0
        g1[1] = (int)(1u << 16);                      // tensor_dim0 = 1
        g1[2] = (int)((unsigned)NF << 16);            // tensor_dim1 = 2048
        g1[3] = (int)(1u << 16);                      // tile_dim0 = 1
        g1[4] = NF;                                   // tile_dim1 = 2048
        g1[5] = NJALL;                                // tensor_dim0_stride = 25 elems
        g1[6] = 0;
        g1[7] = 0;

        v4i z4 = {0, 0, 0, 0};                        // groups 2/3 unused (2D)
        v8i z8 = {0, 0, 0, 0, 0, 0, 0, 0};
        __builtin_amdgcn_tensor_load_to_lds(g0, g1, z4, z4, z8, 0);
        __builtin_amdgcn_s_wait_tensorcnt(0);
    }
    __syncthreads();

    // ---- pass 1: mean ----
    float s = 0.0f;
    for (int k = tid; k < NF; k += TPB) s += xs[k];
    red[tid] = s;
    __syncthreads();
    for (int off = TPB / 2; off > 0; off >>= 1) {
        if (tid < off) red[tid] += red[tid + off];
        __syncthreads();
    }
    if (tid == 0) sBcast[0] = red[0] * (1.0f / (float)NF);
    __syncthreads();
    const float mean = sBcast[0];
    __syncthreads();

    // ---- pass 2: centered 2nd and 4th moments ----
    float s2 = 0.0f, s4 = 0.0f;
    for (int k = tid; k < NF; k += TPB) {
        float d  = xs[k] - mean;
        float d2 = d * d;
        s2 += d2;
        s4 += d2 * d2;
    }
    red[tid] = s2;
    __syncthreads();
    for (int off = TPB / 2; off > 0; off >>= 1) {
        if (tid < off) red[tid] += red[tid + off];
        __syncthreads();
    }
    if (tid == 0) sBcast[1] = red[0];
    __syncthreads();
    red[tid] = s4;
    __syncthreads();
    for (int off = TPB / 2; off > 0; off >>= 1) {
        if (tid < off) red[tid] += red[tid + off];
        __syncthreads();
    }
    if (tid == 0) sBcast[2] = red[0];
    __syncthreads();

    // ---- bitonic sort of 2048 elements in LDS ----
    for (int kk = 2; kk <= NF; kk <<= 1) {
        for (int jj = kk >> 1; jj > 0; jj >>= 1) {
            for (int i = tid; i < NF; i += TPB) {
                int ixj = i ^ jj;
                if (ixj > i) {
                    float a  = xs[i];
                    float bv = xs[ixj];
                    bool up  = ((i & kk) == 0);
                    if (up ? (a > bv) : (a < bv)) { xs[i] = bv; xs[ixj] = a; }
                }
            }
            __syncthreads();
        }
    }

    if (tid == 0) {
        const float sum2 = sBcast[1];
        const float sum4 = sBcast[2];
        const float mn   = xs[0];
        const float mx   = xs[NF - 1];
        // q25: pos = 0.25*2047 = 511.75  -> 0.25*xs[511] + 0.75*xs[512]
        const float q25  = xs[511] * 0.25f + xs[512] * 0.75f;
        const float med  = xs[(NF - 1) / 2];
        // q75: pos = 0.75*2047 = 1535.25 -> 0.75*xs[1535] + 0.25*xs[1536]
        const float q75  = xs[1535] * 0.75f + xs[1536] * 0.25f;
        const float var  = sum2 * (1.0f / (float)(NF - 1));
        const float sd   = sqrtf(var);
        const float kurt = sum4 / ((float)NF * var * var) - 3.0f;

        const float st[8] = { mx, q25, med, q75, mean, sd, mx - mn, kurt };
        #pragma unroll
        for (int si = 0; si < 8; ++si) {
            statA[(((size_t)(sig * 8 + si) * NB) + b) * KDIM + cj] = st[si];
        }
    }
}

// ---------------------------------------------------------------------------
// Kernel 2: fuse GEMM per s-slice with fp32 WMMA.
//   flat2[b][e*16+s] = sum_k statA[s][b][k] * W2[e][k] + b2[e],  k = c*12+j
// One wave (32 threads) per 16x16 (MxN) tile, 48 k-steps of K=4.
// A frag (16x4 f32): lanes 0-15 hold K=0,1; lanes 16-31 hold K=2,3.
// B frag (4x16 f32): mirrored K split across lane halves, N = lane%16.
// C/D (16x16 f32): VGPR r -> M = r + 8*(lane>=16), N = lane%16.
// ---------------------------------------------------------------------------
__global__ void __launch_bounds__(32)
statnet_fuse_wmma_kernel(const float* __restrict__ statA,
                         const float* __restrict__ W2,
                         const float* __restrict__ b2,
                         float* __restrict__ flat2) {
    const int s    = blockIdx.x;     // 0..15
    const int tm   = blockIdx.y;     // 0..3  (b tile)
    const int tn   = blockIdx.z;     // 0..3  (e tile)
    const int lane = threadIdx.x;    // 0..31
    const int half = lane >> 4;
    const int l16  = lane & 15;

    const float* A = statA + (size_t)s * NB * KDIM;
    const int m = tm * 16 + l16;
    const int n = tn * 16 + l16;

    v8f acc = {};
    #pragma unroll 4
    for (int k0 = 0; k0 < KDIM; k0 += 4) {
        const int ka = k0 + half * 2;
        v2f av, bv;
        av.x = A[(size_t)m * KDIM + ka];
        av.y = A[(size_t)m * KDIM + ka + 1];
        bv.x = W2[(size_t)n * KDIM + ka];      // W2[e][c][j] row-major: B[k][n]
        bv.y = W2[(size_t)n * KDIM + ka + 1];
        acc = __builtin_amdgcn_wmma_f32_16x16x4_f32(
            /*neg_a=*/false, av, /*neg_b=*/false, bv,
            /*c_mod=*/(short)0, acc, /*reuse_a=*/false, /*reuse_b=*/false);
    }

    const float bias = b2[n];
    #pragma unroll
    for (int r = 0; r < 8; ++r) {
        const int mrow = tm * 16 + r + half * 8;
        flat2[(size_t)mrow * (NE * NS) + n * NS + s] = acc[r] + bias;
    }
}

// ---------------------------------------------------------------------------
// Kernel 3: logits = flat2 @ Wl.T + bl   (64x1024 . 1024x5 -> 64x5)
// ---------------------------------------------------------------------------
__global__ void __launch_bounds__(32)
statnet_logits_kernel(const float* __restrict__ flat2,
                      const float* __restrict__ Wl,
                      const float* __restrict__ bl,
                      float* __restrict__ out) {
    const int idx = blockIdx.x * 32 + threadIdx.x;
    if (idx >= NB * NCLS) return;
    const int b   = idx / NCLS;
    const int cls = idx - b * NCLS;
    const float* fr = flat2 + (size_t)b * (NE * NS);
    const float* wr = Wl + (size_t)cls * (NE * NS);
    float acc = bl[cls];
    #pragma unroll 8
    for (int k = 0; k < NE * NS; ++k) acc = fmaf(fr[k], wr[k], acc);
    out[(size_t)b * NCLS + cls] = acc;
}

// ---------------------------------------------------------------------------
extern "C" void kernel_launch(void* const* d_in, const int* in_sizes, int n_in,
                              void* d_out, int out_size, void* d_ws, size_t ws_size,
                              hipStream_t stream) {
    // inputs: 0=features 1=W1(dead) 2=b1(dead) 3=W2 4=b2 5=Wl 6=bl
    const float* feat = (const float*)d_in[0];
    const float* W2   = (const float*)d_in[3];
    const float* b2   = (const float*)d_in[4];
    const float* Wl   = (const float*)d_in[5];
    const float* bl   = (const float*)d_in[6];

    float* statA = (float*)d_ws;                       // 16*64*192 f32 = 768 KB
    float* flat2 = statA + (size_t)NS * NB * KDIM;     // 64*1024 f32 = 256 KB

    const int ncols = 2 * NB * NC * NJ;                // 24576 columns
    statnet_stats_kernel<<<ncols, TPB, 0, stream>>>(feat, statA);

    dim3 g(NS, NB / 16, NE / 16);                      // (16, 4, 4)
    statnet_fuse_wmma_kernel<<<g, 32, 0, stream>>>(statA, W2, b2, flat2);

    statnet_logits_kernel<<<(NB * NCLS + 31) / 32, 32, 0, stream>>>(flat2, Wl, bl, (float*)d_out);
}